// StatisticalFeatureExtractor_63797444215020
// MI455X (gfx1250) — compile-verified
//
#include <hip/hip_runtime.h>

typedef __attribute__((ext_vector_type(2))) float v2f;
typedef __attribute__((ext_vector_type(8))) float v8f;

#define T_LEN 65536
#define NTHREADS 256
#define VEC_ITERS (T_LEN / (NTHREADS * 4)) // 64
#define FEPS 1e-8f

__device__ __forceinline__ float sgnf(float v) {
    return (v > 0.0f) ? 1.0f : ((v < 0.0f) ? -1.0f : 0.0f);
}

// Reduce 256 floats in LDS to a scalar using V_WMMA_F32_16X16X4_F32 with B=ones:
// D[m][n] = sum_k A[m][k] + C[m][n].  Sum is permutation-invariant, so the
// lane->(m,k) slot mapping is irrelevant.  Four accumulating WMMAs consume all
// 256 partials; D holds 16 row-sums (rows r in VGPR r lanes 0-15, rows r+8 in
// VGPR r lanes 16-31).  Summing the 8 D VGPRs per lane + shfl_xor(16) yields
// the total in every lane.  Must be called by a full wave (EXEC all ones).
__device__ __forceinline__ float wmma_sum256(const float* arr, int lane) {
    v8f acc = {0.f, 0.f, 0.f, 0.f, 0.f, 0.f, 0.f, 0.f};
#pragma unroll
    for (int i = 0; i < 4; ++i) {
        v2f a, b;
        a.x = arr[i * 64 + 2 * lane];
        a.y = arr[i * 64 + 2 * lane + 1];
        b.x = 1.0f;
        b.y = 1.0f;
        acc = __builtin_amdgcn_wmma_f32_16x16x4_f32(false, a, false, b,
                                                    (short)0, acc, false, false);
    }
    float t = acc[0] + acc[1] + acc[2] + acc[3] + acc[4] + acc[5] + acc[6] + acc[7];
    t += __shfl_xor(t, 16, 32); // combine rows 0-7 (lanes<16) with rows 8-15
    return t;
}

__global__ __launch_bounds__(NTHREADS)
void feat_kernel(const float* __restrict__ x, float* __restrict__ out) {
    const int row = blockIdx.x;
    const int tid = threadIdx.x;
    const float* __restrict__ base = x + (size_t)row * T_LEN;

    __shared__ float sh0[NTHREADS], sh1[NTHREADS], sh2[NTHREADS],
                     sh3[NTHREADS], sh4[NTHREADS];
    __shared__ float shmin[NTHREADS], shmax[NTHREADS];
    __shared__ float s_mean;

    // ---------------- pass 1: raw streaming stats ----------------
    float s1 = 0.f, s2 = 0.f, sa = 0.f, sq = 0.f, zc = 0.f;
    float mn = 3.402823e38f, mx = -3.402823e38f;
#pragma unroll 4
    for (int j = 0; j < VEC_ITERS; ++j) {
        const int idx = (j * NTHREADS + tid) * 4; // coalesced: lanes adjacent
        float4 v = *reinterpret_cast<const float4*>(base + idx);
        float prev = (idx == 0) ? v.x : base[idx - 1]; // L0 hit (neighbor lane's line)
        float ps = sgnf(prev);
        float e0 = v.x, e1 = v.y, e2 = v.z, e3 = v.w;
        float e[4] = {e0, e1, e2, e3};
#pragma unroll
        for (int k = 0; k < 4; ++k) {
            float val = e[k];
            s1 += val;
            s2 = fmaf(val, val, s2);
            float a = fabsf(val);
            sa += a;
            sq += sqrtf(a);
            mn = fminf(mn, val);
            mx = fmaxf(mx, val);
            float s = sgnf(val);
            zc += (s != ps) ? 1.0f : 0.0f;
            ps = s;
        }
    }
    sh0[tid] = s1; sh1[tid] = s2; sh2[tid] = sa; sh3[tid] = sq; sh4[tid] = zc;
    shmin[tid] = mn; shmax[tid] = mx;
    __syncthreads();

    // min/max tree (not expressible as a WMMA sum)
    for (int s = NTHREADS / 2; s > 0; s >>= 1) {
        if (tid < s) {
            shmin[tid] = fminf(shmin[tid], shmin[tid + s]);
            shmax[tid] = fmaxf(shmax[tid], shmax[tid + s]);
        }
        __syncthreads();
    }

    float S1 = 0.f, S2 = 0.f, SA = 0.f, SQ = 0.f, ZC = 0.f;
    if (tid < 32) { // wave 0, full EXEC — WMMA legal
        S1 = wmma_sum256(sh0, tid);
        S2 = wmma_sum256(sh1, tid);
        SA = wmma_sum256(sh2, tid);
        SQ = wmma_sum256(sh3, tid);
        ZC = wmma_sum256(sh4, tid);
        if (tid == 0) s_mean = S1 * (1.0f / (float)T_LEN);
    }
    __syncthreads();
    const float mean = s_mean;

    // ---------------- pass 2: centered moments + MCR (reads hit L2) ----------
    float s3 = 0.f, s4 = 0.f, mc = 0.f;
#pragma unroll 4
    for (int j = 0; j < VEC_ITERS; ++j) {
        const int idx = (j * NTHREADS + tid) * 4;
        float4 v = *reinterpret_cast<const float4*>(base + idx);
        float prev = (idx == 0) ? v.x : base[idx - 1];
        float ps = sgnf(prev - mean);
        float e[4] = {v.x, v.y, v.z, v.w};
#pragma unroll
        for (int k = 0; k < 4; ++k) {
            float c = e[k] - mean;
            float c2 = c * c;
            s3 = fmaf(c2, c, s3);
            s4 = fmaf(c2, c2, s4);
            float s = sgnf(c);
            mc += (s != ps) ? 1.0f : 0.0f;
            ps = s;
        }
    }
    sh0[tid] = s3; sh1[tid] = s4; sh2[tid] = mc;
    __syncthreads();

    if (tid < 32) {
        float S3 = wmma_sum256(sh0, tid);
        float S4 = wmma_sum256(sh1, tid);
        float MC = wmma_sum256(sh2, tid);
        if (tid == 0) {
            const float invT  = 1.0f / (float)T_LEN;
            const float invT1 = 1.0f / (float)(T_LEN - 1);
            float mean_v = S1 * invT;
            float sqm    = S2 * invT;
            float rms    = sqrtf(sqm);
            float varv   = (S2 - (float)T_LEN * mean_v * mean_v) * invT1; // ddof=1
            float stdv   = sqrtf(varv);
            float peak   = shmax[0];
            float pneg   = shmin[0];
            float ptp    = peak - pneg;
            float apk    = fabsf(peak);
            float crest  = apk / (rms + FEPS);
            float mabs   = SA * invT;
            float shape  = rms / (mabs + FEPS);
            float impls  = apk / (mabs + FEPS);
            float sqmean = SQ * invT;
            float clear_ = apk / (sqmean * sqmean + FEPS);
            float margin = apk / (sqmean + FEPS);
            float std2   = stdv * stdv;
            float skew   = (S3 * invT) / (std2 * stdv + FEPS);
            float kurt   = (S4 * invT) / (std2 * std2 + FEPS) - 3.0f;
            float zcr    = ZC * invT1;
            float mcr    = MC * invT1;
            float energy = S2;

            float* o = out + (size_t)row * 17;
            o[0]  = mean_v; o[1]  = stdv;  o[2]  = varv;  o[3]  = rms;
            o[4]  = peak;   o[5]  = pneg;  o[6]  = ptp;   o[7]  = crest;
            o[8]  = shape;  o[9]  = impls; o[10] = clear_; o[11] = skew;
            o[12] = kurt;   o[13] = zcr;   o[14] = mcr;    o[15] = margin;
            o[16] = energy;
        }
    }
}

extern "C" void kernel_launch(void* const* d_in, const int* in_sizes, int n_in,
                              void* d_out, int out_size, void* d_ws, size_t ws_size,
                              hipStream_t stream) {
    const float* x = (const float*)d_in[0];
    float* out = (float*)d_out;
    const int rows = in_sizes[0] / T_LEN; // 64*16 = 1024
    feat_kernel<<<dim3(rows), dim3(NTHREADS), 0, stream>>>(x, out);
}